// AVWGCN_17514876633689
// MI455X (gfx1250) — compile-verified
//
#include <hip/hip_runtime.h>
#include <math.h>

typedef __attribute__((ext_vector_type(2))) float v2f;
typedef __attribute__((ext_vector_type(4))) float v4f;
typedef __attribute__((ext_vector_type(8))) float v8f;

#define NNODES 10000
#define CDIM   64
#define HID    64
#define BATCH  32
#define EPAD   65          // sE row stride (words) -> conflict-free row walks
#define FS     66          // fragment slot stride (33 lane-pairs * 2 words)

// Fragment-native LDS layout for V_WMMA_F32_16X16X4_F32 (wave32):
//   slot s = (tileIdx*16 + kt), word addr = s*FS + lane*2 + w
//   A: lane<16 -> row=lane,   K=kt*4+{0,1}; lane>=16 -> row=lane-16, K=kt*4+{2,3}
//   B: lane<16 -> col=lane,   K=kt*4+{0,1}; lane>=16 -> col=lane-16, K=kt*4+{2,3}
// Loads are *(v2f*) at even word offsets: banks 2*lane,2*lane+1 => conflict-free.

__global__ __launch_bounds__(256)
void avwgcn_wmma_kernel(const float* __restrict__ x,
                        const float* __restrict__ emb,
                        const float* __restrict__ wpool,
                        const float* __restrict__ bias,
                        float* __restrict__ out)
{
    __shared__ __align__(16) float sE  [CDIM * EPAD];  // relu(E), padded rows (16.6KB)
    __shared__ __align__(16) float sSf [64 * FS];      // softmax(S), B-frag layout (16.9KB)
    __shared__ __align__(16) float sXf [32 * FS];      // X_n, A-frag layout (8.4KB)
    __shared__ __align__(16) float sXGf[32 * FS];      // X_n @ S_n, A-frag layout (8.4KB)

    const int n    = blockIdx.x;
    const int tid  = threadIdx.x;
    const int lane = tid & 31;
    const int wave = tid >> 5;

    const float* E  = emb   + (size_t)n * (CDIM * CDIM);
    const float* W0 = wpool + (size_t)n * (CDIM * HID);
    const float* W1 = wpool + ((size_t)NNODES + n) * (CDIM * HID);

    // Prefetch this node's 32KB of weights (global_prefetch_b8) behind staging/softmax.
    {
        const char* base = (const char*)((tid < 128) ? W0 : W1);
        __builtin_prefetch(base + (size_t)(tid & 127) * 128, 0, 3);
    }

    // ---- Stage relu(E) into padded sE. Banks: (r + c + j) % 64, conflict-free.
    #pragma unroll
    for (int i = 0; i < 4; ++i) {
        const int f = tid + i * 256;               // float4 index
        const int r = f >> 4, c = (f & 15) << 2;
        v4f v = *(const v4f*)(E + (size_t)f * 4);
        float* d = sE + r * EPAD + c;
        d[0] = fmaxf(v.x, 0.0f);
        d[1] = fmaxf(v.y, 0.0f);
        d[2] = fmaxf(v.z, 0.0f);
        d[3] = fmaxf(v.w, 0.0f);
    }

    // ---- Stage X_n directly into A-fragment layout (two b64 stores per float4).
    #pragma unroll
    for (int i = 0; i < 2; ++i) {
        const int f  = tid + i * 256;              // float4 index
        const int b  = f >> 4, c4 = f & 15;        // batch row, kt
        v4f v = *(const v4f*)(x + ((size_t)b * NNODES + n) * CDIM + (size_t)c4 * 4);
        const int mi = b >> 4, bl = b & 15;
        float* slot = sXf + (size_t)(mi * 16 + c4) * FS;
        v2f lo, hi;
        lo.x = v.x; lo.y = v.y;                    // K = 4*kt + {0,1} -> lane bl
        hi.x = v.z; hi.y = v.w;                    // K = 4*kt + {2,3} -> lane bl+16
        *(v2f*)(slot + bl * 2)        = lo;
        *(v2f*)(slot + (bl + 16) * 2) = hi;
    }
    __syncthreads();

    // ---- Row softmax of relu(E); results written straight into B-frag layout.
    if (tid < CDIM) {
        const float* row = sE + tid * EPAD;
        float mx = 0.0f;                            // relu'd values are >= 0
        for (int j = 0; j < CDIM; ++j) mx = fmaxf(mx, row[j]);

        const int kt = tid >> 2;                    // K-tile of this S-row
        const int hi = (tid & 2) ? 32 : 0;          // upper half-wave word offset
        const int w  = tid & 1;                     // word within lane pair
        float sum = 0.0f;
        for (int j = 0; j < CDIM; ++j) {
            const float ev = __expf(row[j] - mx);
            sum += ev;
            sSf[(size_t)((j >> 4) * 16 + kt) * FS + ((j & 15) * 2) + hi + w] = ev;
        }
        const float inv = 1.0f / sum;
        for (int j = 0; j < CDIM; ++j)
            sSf[(size_t)((j >> 4) * 16 + kt) * FS + ((j & 15) * 2) + hi + w] *= inv;
    }
    __syncthreads();

    // ---- WMMA tiling: each wave owns one 16x16 tile of the 32x64 outputs.
    const int mi   = wave >> 2;                     // batch tile 0..1
    const int ni   = wave & 3;                      // hid tile 0..3
    const int colL = lane & 15;
    const int col  = ni * 16 + colL;
    const int koff = (lane >> 4) << 1;              // upper lanes: K+2
    const int r0   = mi * 16 + ((lane >> 4) << 3);  // C/D row base

    const float* aBase  = sXf  + (size_t)(mi * 16) * FS + lane * 2;
    const float* bBase  = sSf  + (size_t)(ni * 16) * FS + lane * 2;
    const float* agBase = sXGf + (size_t)(mi * 16) * FS + lane * 2;

    // XG = X @ S  (conflict-free b64 fragment loads)
    v8f acc = {};
    #pragma unroll
    for (int kt = 0; kt < 16; ++kt) {
        v2f a = *(const v2f*)(aBase + (size_t)kt * FS);
        v2f b = *(const v2f*)(bBase + (size_t)kt * FS);
        acc = __builtin_amdgcn_wmma_f32_16x16x4_f32(false, a, false, b,
                                                    (short)0, acc, false, false);
    }

    // Spill XG tile directly in A-fragment layout (conflict-free scalar stores).
    {
        const int ktw = ni * 4 + (colL >> 2);       // K-tile this column lands in
        const int dw  = ((colL & 2) ? 32 : 0) + (colL & 1);
        float* dst = sXGf + (size_t)(mi * 16 + ktw) * FS + dw;
        #pragma unroll
        for (int r = 0; r < 8; ++r)
            dst[((r0 + r) & 15) * 2] = acc[r];
    }
    __syncthreads();

    // Out = X @ W0 + XG @ W1 (+bias); weights streamed from global (prefetched).
    v8f o = {};
    #pragma unroll 4
    for (int kt = 0; kt < 16; ++kt) {
        const int k = kt * 4 + koff;
        v2f a0 = *(const v2f*)(aBase + (size_t)kt * FS);
        v2f b0;
        b0.x = W0[(size_t)(k)     * HID + col];
        b0.y = W0[(size_t)(k + 1) * HID + col];
        o = __builtin_amdgcn_wmma_f32_16x16x4_f32(false, a0, false, b0,
                                                  (short)0, o, false, false);
        v2f a1 = *(const v2f*)(agBase + (size_t)kt * FS);
        v2f b1;
        b1.x = W1[(size_t)(k)     * HID + col];
        b1.y = W1[(size_t)(k + 1) * HID + col];
        o = __builtin_amdgcn_wmma_f32_16x16x4_f32(false, a1, false, b1,
                                                  (short)0, o, false, false);
    }

    const float bv = bias[(size_t)n * HID + col];
    #pragma unroll
    for (int r = 0; r < 8; ++r) {
        const int b = r0 + r;                       // batch index
        out[((size_t)b * NNODES + n) * HID + col] = o[r] + bv;
    }
}

extern "C" void kernel_launch(void* const* d_in, const int* in_sizes, int n_in,
                              void* d_out, int out_size, void* d_ws, size_t ws_size,
                              hipStream_t stream) {
    const float* x     = (const float*)d_in[0];
    const float* emb   = (const float*)d_in[1];
    const float* wpool = (const float*)d_in[2];
    const float* bias  = (const float*)d_in[3];
    float*       out   = (float*)d_out;
    (void)in_sizes; (void)n_in; (void)out_size; (void)d_ws; (void)ws_size;

    avwgcn_wmma_kernel<<<NNODES, 256, 0, stream>>>(x, emb, wpool, bias, out);
}